// PQC6QDummy3_74345883893942
// MI455X (gfx1250) — compile-verified
//
#include <hip/hip_runtime.h>

// 6-qubit PQC statevector simulator for MI455X (gfx1250).
// One wave32, state resident in LDS, gates executed as
// V_WMMA_F32_16X16X4_F32: S_new(4x16) = Gate(4x4 complex) @ S(4x16 complex),
// complex product expanded into 4 real f32 WMMAs per gate.
// All 24 cos/sin(theta/2) computed in parallel across lanes up front with
// hardware v_cos_f32/v_sin_f32 (avoids 24x serialized scalar Payne-Hanek).

typedef float v2f __attribute__((ext_vector_type(2)));
typedef float v8f __attribute__((ext_vector_type(8)));

// Map (m = gated-pair value bits, n = remaining-4-qubit bits) back to the
// flat amplitude index for gate on qubit axes (a, b) of the (2,)*6 tensor.
// Axis 0 is the most significant index bit (row-major). moveaxis((a,b)->(0,1))
// means m = q_a*2 + q_b and n packs the remaining axes in increasing axis
// order, MSB first.
__device__ __forceinline__ int amp_index(int m, int n, int a, int b) {
  int idx = 0;
  int nb = 3;
#pragma unroll
  for (int ax = 0; ax < 6; ++ax) {
    int bit;
    if (ax == a)      bit = (m >> 1) & 1;
    else if (ax == b) bit = m & 1;
    else            { bit = (n >> nb) & 1; --nb; }
    idx = (idx << 1) | bit;
  }
  return idx;
}

__global__ __launch_bounds__(32)
void pqc6q_wmma_kernel(const float* __restrict__ params, float* __restrict__ out) {
  __shared__ float s_re[64];
  __shared__ float s_im[64];
  __shared__ float s_cos[24];
  __shared__ float s_sin[24];

  const int L = threadIdx.x;  // 0..31, single wave32

  // |000000> initial state
  s_re[L]      = 0.0f; s_im[L]      = 0.0f;
  s_re[L + 32] = 0.0f; s_im[L + 32] = 0.0f;
  if (L == 0) s_re[0] = 1.0f;

  // All 24 gate angles processed in parallel: one v_cos_f32 + one v_sin_f32
  // total (hardware trans ops) instead of 24 serialized scalar sin/cos.
  if (L < 24) {
    const float t = 0.5f * params[L];
    s_cos[L] = __cosf(t);
    s_sin[L] = __sinf(t);
  }
  __syncthreads();

  // Fragment coordinates for this lane:
  // A (16x4, MxK): lane holds M = L%16; VGPR j holds K = kBase + j
  // B (4x16, KxN): lane holds N = L%16; VGPR j holds K = kBase + j
  const int nIdx  = L & 15;
  const int kBase = (L < 16) ? 0 : 2;

#pragma unroll
  for (int gi = 0; gi < 24; ++gi) {
    const int g  = gi % 6;
    const int qa = g;
    const int qb = (g + 1) % 6;                 // ring pairs (g, g+1 mod 6)
    const bool is_rzx = (((gi / 6) & 1) == 0);  // per block: 6x RZX then 6x RXX

    const float ct = s_cos[gi];
    const float st = s_sin[gi];

    // Gate matrix entries (rows >= 4 are implicitly zero: patterns below
    // only fire for M < 4 when K < 4).
    auto g_re = [&](int M, int K) -> float {
      return (M == K) ? ct : 0.0f;
    };
    auto g_im = [&](int M, int K) -> float {
      if (is_rzx) // -i sin * ZX : ZX[M][K] = +-1 when M^K==1, sign by M&2
        return ((M ^ K) == 1) ? ((M & 2) ? st : -st) : 0.0f;
      else        // -i sin * XX : XX[M][K] = 1 when M^K==3
        return ((M ^ K) == 3) ? -st : 0.0f;
    };

    const int M = nIdx;
    v2f a_re  = { g_re(M, kBase), g_re(M, kBase + 1) };
    v2f a_im  = { g_im(M, kBase), g_im(M, kBase + 1) };
    v2f a_imn = { -a_im.x, -a_im.y };  // f32 WMMA has no A-neg modifier

    // Gather B fragment (state reshaped 4x16 for this qubit pair) from LDS.
    const int i0 = amp_index(kBase,     nIdx, qa, qb);
    const int i1 = amp_index(kBase + 1, nIdx, qa, qb);
    v2f b_re = { s_re[i0], s_re[i1] };
    v2f b_im = { s_im[i0], s_im[i1] };

    v8f z = {};
    // Re = Ar*Br - Ai*Bi ; Im = Ar*Bi + Ai*Br
    v8f d_re = __builtin_amdgcn_wmma_f32_16x16x4_f32(
        false, a_re,  false, b_re, (short)0, z,    false, false);
    d_re     = __builtin_amdgcn_wmma_f32_16x16x4_f32(
        false, a_imn, false, b_im, (short)0, d_re, false, false);
    v8f d_im = __builtin_amdgcn_wmma_f32_16x16x4_f32(
        false, a_re,  false, b_im, (short)0, z,    false, false);
    d_im     = __builtin_amdgcn_wmma_f32_16x16x4_f32(
        false, a_im,  false, b_re, (short)0, d_im, false, false);

    __syncthreads();  // S_NOP for 1-wave WG; acts as compiler LDS fence
    // D 16x16 f32 layout: lanes 0-15 hold rows 0-7 in d[0..7]; the live
    // result rows are 0..3 (rest are zero-padded).
    if (L < 16) {
#pragma unroll
      for (int v = 0; v < 4; ++v) {
        const int o = amp_index(v, L, qa, qb);
        s_re[o] = d_re[v];
        s_im[o] = d_im[v];
      }
    }
    __syncthreads();
  }

  // Emit complex64 state as interleaved (re, im) float pairs: 128 floats.
  out[2 * L + 0]        = s_re[L];
  out[2 * L + 1]        = s_im[L];
  out[2 * (L + 32) + 0] = s_re[L + 32];
  out[2 * (L + 32) + 1] = s_im[L + 32];
}

extern "C" void kernel_launch(void* const* d_in, const int* in_sizes, int n_in,
                              void* d_out, int out_size, void* d_ws, size_t ws_size,
                              hipStream_t stream) {
  (void)in_sizes; (void)n_in; (void)out_size; (void)d_ws; (void)ws_size;
  const float* params = (const float*)d_in[0];  // 24 angles, float32
  float* out = (float*)d_out;                   // 64 complex64 -> 128 floats
  pqc6q_wmma_kernel<<<dim3(1), dim3(32), 0, stream>>>(params, out);
}